// SoftDetect_19043884990743
// MI455X (gfx1250) — compile-verified
//
#include <hip/hip_runtime.h>
#include <stdint.h>

// ---------------- problem constants (match reference) ----------------
#define HH   512
#define WW   640
#define BB   8
#define HWSZ (HH * WW)
#define TOPK 4096
#define CAPL 65536   // per-image compacted candidate cap
#define CAPS 8192    // per-image selected cap (pow2 for bitonic sort)

typedef __attribute__((ext_vector_type(2))) float v2f;
typedef __attribute__((ext_vector_type(8))) float v8f;

// ============================================================
// Stage 1: iterated 5x5 NMS (SAME padding with -inf), memory bound;
// whole 10.5MB map lives in the 192MB L2, so 25-tap reads are cheap.
// Interior pixels take an unchecked constant-offset fast path.
// ============================================================
__global__ void k_mask0(const float* __restrict__ s, uint8_t* __restrict__ mask) {
    int t = blockIdx.x * blockDim.x + threadIdx.x;
    if (t >= BB * HWSZ) return;
    int b = t / HWSZ, p = t - b * HWSZ, y = p / WW, x = p - y * WW;
    float v = s[t], mx = -INFINITY;
    if (y >= 2 && y < HH - 2 && x >= 2 && x < WW - 2) {
        const float* c = s + t;
        #pragma unroll
        for (int dy = -2; dy <= 2; ++dy)
            #pragma unroll
            for (int dx = -2; dx <= 2; ++dx)
                mx = fmaxf(mx, c[dy * WW + dx]);
    } else {
        for (int dy = -2; dy <= 2; ++dy)
            for (int dx = -2; dx <= 2; ++dx) {
                int yy = y + dy, xx = x + dx;
                if (yy >= 0 && yy < HH && xx >= 0 && xx < WW)
                    mx = fmaxf(mx, s[b * HWSZ + yy * WW + xx]);
            }
    }
    mask[t] = (v == mx) ? 1 : 0;
}

__global__ void k_supp(const float* __restrict__ s, const uint8_t* __restrict__ mask,
                       uint8_t* __restrict__ supp, float* __restrict__ ss) {
    int t = blockIdx.x * blockDim.x + threadIdx.x;
    if (t >= BB * HWSZ) return;
    int b = t / HWSZ, p = t - b * HWSZ, y = p / WW, x = p - y * WW;
    int any = 0;
    if (y >= 2 && y < HH - 2 && x >= 2 && x < WW - 2) {
        const uint8_t* c = mask + t;
        #pragma unroll
        for (int dy = -2; dy <= 2; ++dy)
            #pragma unroll
            for (int dx = -2; dx <= 2; ++dx)
                any |= c[dy * WW + dx];
    } else {
        for (int dy = -2; dy <= 2; ++dy)
            for (int dx = -2; dx <= 2; ++dx) {
                int yy = y + dy, xx = x + dx;
                if (yy >= 0 && yy < HH && xx >= 0 && xx < WW)
                    any |= mask[b * HWSZ + yy * WW + xx];
            }
    }
    supp[t] = (uint8_t)any;
    ss[t] = any ? 0.0f : s[t];
}

__global__ void k_newmax(const float* __restrict__ ss, const uint8_t* __restrict__ supp,
                         uint8_t* __restrict__ mask) {
    int t = blockIdx.x * blockDim.x + threadIdx.x;
    if (t >= BB * HWSZ) return;
    int b = t / HWSZ, p = t - b * HWSZ, y = p / WW, x = p - y * WW;
    float mx = -INFINITY;
    if (y >= 2 && y < HH - 2 && x >= 2 && x < WW - 2) {
        const float* c = ss + t;
        #pragma unroll
        for (int dy = -2; dy <= 2; ++dy)
            #pragma unroll
            for (int dx = -2; dx <= 2; ++dx)
                mx = fmaxf(mx, c[dy * WW + dx]);
    } else {
        for (int dy = -2; dy <= 2; ++dy)
            for (int dx = -2; dx <= 2; ++dx) {
                int yy = y + dy, xx = x + dx;
                if (yy >= 0 && yy < HH && xx >= 0 && xx < WW)
                    mx = fmaxf(mx, ss[b * HWSZ + yy * WW + xx]);
            }
    }
    if ((ss[t] == mx) && !supp[t]) mask[t] = 1;   // max_mask |= new_max & ~supp
}

// ============================================================
// Stage 2: exact top-4096 per image, sorted (desc score, asc index).
// key = (~float_bits(score) << 32) | pixel_index  -> sort ascending.
// ============================================================
__global__ void k_zero(uint32_t* cnt) {
    if (blockIdx.x == 0 && threadIdx.x < 16) cnt[threadIdx.x] = 0;
}

__global__ void k_compact(const float* __restrict__ s, const uint8_t* __restrict__ mask,
                          uint64_t* __restrict__ keys, uint32_t* __restrict__ cnt) {
    int t = blockIdx.x * blockDim.x + threadIdx.x;
    if (t >= BB * HWSZ) return;
    if (!mask[t]) return;
    int b = t / HWSZ, p = t - b * HWSZ, y = p / WW, x = p - y * WW;
    // border zeroing: rows/cols [0..r] and [H-r..) with r=2
    if (y < 3 || x < 3 || y >= HH - 2 || x >= WW - 2) return;
    float v = s[t];
    if (!(v > 0.0f)) return;
    uint32_t pos = atomicAdd((unsigned int*)&cnt[b], 1u);
    if (pos < CAPL)
        keys[(size_t)b * CAPL + pos] =
            ((uint64_t)(~__float_as_uint(v)) << 32) | (uint32_t)p;
}

// one block per image: radix-select 4096th smallest high-word, gather, pad
__global__ void k_select(const uint64_t* __restrict__ keys, const uint32_t* __restrict__ cnt,
                         uint64_t* __restrict__ sel) {
    __shared__ int hist[256];
    __shared__ uint32_t s_prefix, s_maskb;
    __shared__ int s_kth, s_gcnt;
    int b = blockIdx.x;
    const uint64_t* kb = keys + (size_t)b * CAPL;
    int n = (int)cnt[b]; if (n > CAPL) n = CAPL;

    uint32_t thr = 0xFFFFFFFFu;
    if (n > TOPK) {
        if (threadIdx.x == 0) { s_prefix = 0; s_maskb = 0; s_kth = TOPK; }
        __syncthreads();
        for (int shift = 24; shift >= 0; shift -= 8) {
            for (int i = threadIdx.x; i < 256; i += blockDim.x) hist[i] = 0;
            __syncthreads();
            uint32_t pf = s_prefix, mk = s_maskb;
            for (int i = threadIdx.x; i < n; i += blockDim.x) {
                uint32_t k32 = (uint32_t)(kb[i] >> 32);
                if ((k32 & mk) == pf) atomicAdd(&hist[(k32 >> shift) & 0xFF], 1);
            }
            __syncthreads();
            if (threadIdx.x == 0) {
                int cum = 0, chosen = 255;
                for (int bin = 0; bin < 256; ++bin) {
                    int c = hist[bin];
                    if (cum + c >= s_kth) { chosen = bin; break; }
                    cum += c;
                }
                s_kth -= cum;
                s_prefix |= (uint32_t)chosen << shift;
                s_maskb |= 0xFFu << shift;
            }
            __syncthreads();
        }
        thr = s_prefix;
    }
    if (threadIdx.x == 0) s_gcnt = 0;
    __syncthreads();
    for (int i = threadIdx.x; i < n; i += blockDim.x) {
        uint64_t k = kb[i];
        if ((uint32_t)(k >> 32) <= thr) {
            int pos = atomicAdd(&s_gcnt, 1);
            if (pos < CAPS) sel[(size_t)b * CAPS + pos] = k;
        }
    }
    __syncthreads();
    int g = s_gcnt; if (g > CAPS) g = CAPS;
    // pad: score-0 keys with ascending small indices (sort after all reals)
    for (int i = g + threadIdx.x; i < CAPS; i += blockDim.x)
        sel[(size_t)b * CAPS + i] = (0xFFFFFFFFull << 32) | (uint32_t)(i - g);
}

// bitonic sort of 8192 u64 keys in 64KB LDS, one block per image
__global__ __launch_bounds__(512) void k_sort(const uint64_t* __restrict__ sel,
                                              int* __restrict__ topidx) {
    __shared__ uint64_t sk[CAPS];   // 64KB (CDNA5 WGP has 320KB LDS)
    int b = blockIdx.x;
    for (int i = threadIdx.x; i < CAPS; i += blockDim.x) sk[i] = sel[(size_t)b * CAPS + i];
    __syncthreads();
    for (int k = 2; k <= CAPS; k <<= 1)
        for (int j = k >> 1; j > 0; j >>= 1) {
            for (int i = threadIdx.x; i < CAPS; i += blockDim.x) {
                int ixj = i ^ j;
                if (ixj > i) {
                    uint64_t a = sk[i], c = sk[ixj];
                    if (((i & k) == 0) ? (a > c) : (a < c)) { sk[i] = c; sk[ixj] = a; }
                }
            }
            __syncthreads();
        }
    for (int i = threadIdx.x; i < TOPK; i += blockDim.x)
        topidx[b * TOPK + i] = (int)(uint32_t)(sk[i] & 0xFFFFFFFFu);
}

// ============================================================
// Stage 3: per-keypoint soft-argmax via V_WMMA_F32_16X16X4_F32.
//   A = x_exp[16 kp x 28]   (K padded 25->28, 7 chained WMMAs)
//   B = G[28 x 16], cols: [1, gx, gy, gx^2+gy^2, 0...]
// Closed form: x=S1/S0, y=S2/S0, disp=(S3/S0 - x^2 - y^2)/r^2.
// All lane-dependent constants are compile-time pairs selected by `lo`
// (single v_cndmask each) to avoid divergent constant materialization.
// ============================================================
__global__ __launch_bounds__(256) void k_softdetect(const float* __restrict__ s,
                                                    const int* __restrict__ topidx,
                                                    float* __restrict__ out) {
    __shared__ float tile[8][16][16];            // per-wave D spill (8KB)
    int lane = threadIdx.x & 31, wv = threadIdx.x >> 5;
    int g = blockIdx.x * 8 + wv;                 // 256 waves per image
    int b = g >> 8;
    int n0 = (g & 255) << 4;                     // first of 16 keypoints for wave
    bool lo = (lane < 16);
    int m = lane & 15;                           // keypoint row / output column
    int n = n0 + m;
    int idx = topidx[b * TOPK + n];
    int cy = idx / WW, cx = idx - cy * WW;
    const float* sb = s + b * HWSZ;

    // -------- gather patch values (A layout: VGPR0={K0|K2}, VGPR1={K1|K3}) ----
    float vals[14];
    float vmax = -INFINITY;
    bool inY = (cy >= 2) & (cy < HH - 2);
    bool inX = (cx >= 2) & (cx < WW - 2);
    bool interior = inY & inX;                   // true for all real keypoints
    #pragma unroll
    for (int c = 0; c < 7; ++c) {
        #pragma unroll
        for (int t = 0; t < 2; ++t) {
            const int Kl = 4 * c + t;            // compile-time
            const int Kh = 4 * c + 2 + t;        // compile-time
            const bool vL = (Kl < 25), vH = (Kh < 25);
            const int oL = vL ? (Kl / 5 - 2) * WW + (Kl % 5 - 2) : 0;
            const int oH = vH ? (Kh / 5 - 2) * WW + (Kh % 5 - 2) : 0;
            const int dyL = vL ? Kl / 5 - 2 : 0, dxL = vL ? Kl % 5 - 2 : 0;
            const int dyH = vH ? Kh / 5 - 2 : 0, dxH = vH ? Kh % 5 - 2 : 0;
            bool valid = lo ? vL : vH;
            int off = lo ? oL : oH;
            float v = 0.0f;
            if (valid) {
                if (interior) {
                    v = sb[idx + off];
                } else {
                    int yy = cy + (lo ? dyL : dyH);
                    int xx = cx + (lo ? dxL : dxH);
                    if (yy >= 0 && yy < HH && xx >= 0 && xx < WW)
                        v = sb[yy * WW + xx];    // zero-padded unfold
                }
                vmax = fmaxf(vmax, v);           // max includes zero pads
            }
            vals[2 * c + t] = v;
        }
    }
    vmax = fmaxf(vmax, __shfl_xor(vmax, 16));    // combine lane-pair halves

    // -------- 7 chained WMMAs ------------------------------------------------
    v8f acc = {0.f, 0.f, 0.f, 0.f, 0.f, 0.f, 0.f, 0.f};
    bool m0 = (m == 0), m1 = (m == 1), m2 = (m == 2), m3 = (m == 3);
    #pragma unroll
    for (int c = 0; c < 7; ++c) {
        v2f a, bc;
        #pragma unroll
        for (int t = 0; t < 2; ++t) {
            const int Kl = 4 * c + t, Kh = 4 * c + 2 + t;
            const bool vL = (Kl < 25), vH = (Kh < 25);
            // compile-time G entries for both lane halves (0 when K invalid)
            const float oneL = vL ? 1.0f : 0.0f, oneH = vH ? 1.0f : 0.0f;
            const float gxL = vL ? (float)(Kl % 5 - 2) : 0.0f;
            const float gxH = vH ? (float)(Kh % 5 - 2) : 0.0f;
            const float gyL = vL ? (float)(Kl / 5 - 2) : 0.0f;
            const float gyH = vH ? (float)(Kh / 5 - 2) : 0.0f;
            const float gqL = gxL * gxL + gyL * gyL;
            const float gqH = gxH * gxH + gyH * gyH;
            bool valid = lo ? vL : vH;
            float w = valid ? expf((vals[2 * c + t] - vmax) * 10.0f) : 0.0f;
            float c0 = lo ? oneL : oneH;
            float c1 = lo ? gxL : gxH;
            float c2 = lo ? gyL : gyH;
            float c3 = lo ? gqL : gqH;
            float bv = 0.0f;
            bv = m3 ? c3 : bv;
            bv = m2 ? c2 : bv;
            bv = m1 ? c1 : bv;
            bv = m0 ? c0 : bv;
            if (t == 0) { a.x = w; bc.x = bv; } else { a.y = w; bc.y = bv; }
        }
        acc = __builtin_amdgcn_wmma_f32_16x16x4_f32(
            false, a, false, bc, (short)0, acc, false, false);
    }

    // D layout: VGPR j -> (M=j, N=lane) lanes 0-15, (M=j+8, N=lane-16) else
    int col = lo ? lane : lane - 16;
    int rbase = lo ? 0 : 8;
    #pragma unroll
    for (int j = 0; j < 8; ++j) tile[wv][rbase + j][col] = acc[j];
    __syncthreads();

    if (lo) {
        float S0 = tile[wv][m][0], S1 = tile[wv][m][1];
        float S2 = tile[wv][m][2], S3 = tile[wv][m][3];
        float xr = S1 / S0, yr = S2 / S0;
        float kx = ((float)cx + xr) / (float)(WW - 1) * 2.0f - 1.0f;
        float ky = ((float)cy + yr) / (float)(HH - 1) * 2.0f - 1.0f;
        float disp = (S3 / S0 - xr * xr - yr * yr) * 0.25f;   // r^2 = 4

        // bilinear sample, align_corners=True, reference-matching clipping
        float px = (kx + 1.0f) * 0.5f * (float)(WW - 1);
        float py = (ky + 1.0f) * 0.5f * (float)(HH - 1);
        float x0 = floorf(px), y0 = floorf(py);
        float wx = px - x0, wy = py - y0;
        int x0i = min(max((int)x0, 0), WW - 1);
        int x1i = min(max((int)x0 + 1, 0), WW - 1);
        int y0i = min(max((int)y0, 0), HH - 1);
        int y1i = min(max((int)y0 + 1, 0), HH - 1);
        float v00 = sb[y0i * WW + x0i], v01 = sb[y0i * WW + x1i];
        float v10 = sb[y1i * WW + x0i], v11 = sb[y1i * WW + x1i];
        float sc = v00 * (1 - wx) * (1 - wy) + v01 * wx * (1 - wy)
                 + v10 * (1 - wx) * wy + v11 * wx * wy;

        int o = b * TOPK + n;
        out[o * 2 + 0] = kx;
        out[o * 2 + 1] = ky;
        out[BB * TOPK * 2 + o] = sc;     // kptscore
        out[BB * TOPK * 3 + o] = disp;   // disp
    }
}

// ============================================================
extern "C" void kernel_launch(void* const* d_in, const int* in_sizes, int n_in,
                              void* d_out, int out_size, void* d_ws, size_t ws_size,
                              hipStream_t stream) {
    (void)in_sizes; (void)n_in; (void)out_size; (void)ws_size;
    const float* s = (const float*)d_in[0];
    float* out = (float*)d_out;

    char* w = (char*)d_ws;
    size_t off = 0;
    auto alloc = [&](size_t bytes) -> void* {
        off = (off + 255) & ~(size_t)255;
        void* p = w + off;
        off += bytes;
        return p;
    };
    uint8_t*  mask   = (uint8_t*)alloc((size_t)BB * HWSZ);
    uint8_t*  supp   = (uint8_t*)alloc((size_t)BB * HWSZ);
    float*    ssb    = (float*)alloc((size_t)BB * HWSZ * 4);
    uint64_t* keys   = (uint64_t*)alloc((size_t)BB * CAPL * 8);
    uint32_t* cnt    = (uint32_t*)alloc(256);
    uint64_t* sel    = (uint64_t*)alloc((size_t)BB * CAPS * 8);
    int*      topidx = (int*)alloc((size_t)BB * TOPK * 4);

    int total = BB * HWSZ;
    dim3 blk(256), grd((total + 255) / 256);

    k_mask0<<<grd, blk, 0, stream>>>(s, mask);
    for (int it = 0; it < 2; ++it) {
        k_supp<<<grd, blk, 0, stream>>>(s, mask, supp, ssb);
        k_newmax<<<grd, blk, 0, stream>>>(ssb, supp, mask);
    }
    k_zero<<<1, 64, 0, stream>>>(cnt);
    k_compact<<<grd, blk, 0, stream>>>(s, mask, keys, cnt);
    k_select<<<BB, 256, 0, stream>>>(keys, cnt, sel);
    k_sort<<<BB, 512, 0, stream>>>(sel, topidx);
    k_softdetect<<<(BB * TOPK) / 128, 256, 0, stream>>>(s, topidx, out);
}